// MoDRouter_40329742909554
// MI455X (gfx1250) — compile-verified
//
#include <hip/hip_runtime.h>

typedef __attribute__((ext_vector_type(2))) float v2f;
typedef __attribute__((ext_vector_type(8))) float v8f;

constexpr int B = 4;
constexpr int T = 8192;
constexpr int D = 2048;
constexpr int K = 4096;  // k = ceil(0.5 * T)

// ---------------------------------------------------------------------------
// Kernel 1: router scores via V_WMMA_F32_16X16X4_F32 (full fp32 precision).
// One wave32 computes 16 token scores. A = 16 token rows x 4 d-elems,
// B = gate_w chunk replicated across all 16 columns, C accumulates over D.
// Per ISA layouts, both A and B want per-lane float2 at offset
// (lane<16 ? 0 : 2) within each 4-wide K chunk -> contiguous 8B loads.
// ---------------------------------------------------------------------------
__global__ void __launch_bounds__(256) mod_scores_wmma(
    const float* __restrict__ x, const float* __restrict__ w,
    float* __restrict__ scores)
{
  const int lane = threadIdx.x & 31;
  const int wave = threadIdx.x >> 5;
  const int t0   = (blockIdx.x * 8 + wave) * 16;   // flat token base (b*T + t)
  const int half = lane >> 4;                       // 0: K={0,1}, 1: K={2,3}
  const int m    = lane & 15;                       // token-in-tile (A row)
  const float* row = x + (size_t)(t0 + m) * D + half * 2;
  const float* wp  = w + half * 2;

  v8f c = {};
#pragma unroll 4
  for (int k = 0; k < D; k += 4) {
    v2f a = *(const v2f*)(row + k);
    v2f b = *(const v2f*)(wp + k);
    // D = A x B + C : every column n of D holds score[t0+m]
    c = __builtin_amdgcn_wmma_f32_16x16x4_f32(false, a, false, b,
                                              (short)0, c, false, false);
  }
  // C/D layout: VGPR r, lanes 0-15 -> M=r, lanes 16-31 -> M=r+8.
  // Lane 0 holds scores t0..t0+7, lane 16 holds t0+8..t0+15.
  if (lane == 0 || lane == 16) {
    float* p = scores + t0 + half * 8;
    *(float4*)(p)     = make_float4(c[0], c[1], c[2], c[3]);
    *(float4*)(p + 4) = make_float4(c[4], c[5], c[6], c[7]);
  }
}

// ---------------------------------------------------------------------------
// Kernel 2: exact top-k by rank counting (matches jax.lax.top_k ordering:
// values descending, ties broken by lower index). Scores for one batch sit
// in 32 KB of LDS; each thread ranks one token against all T.
// ---------------------------------------------------------------------------
constexpr int TPB_TOPK = 256;

__global__ void __launch_bounds__(TPB_TOPK) mod_topk_rank(
    const float* __restrict__ scores, int* __restrict__ idx_out)
{
  __shared__ float s[T];                      // 32 KB of the 320 KB WGP LDS
  const int blocksPerBatch = T / TPB_TOPK;    // 32
  const int b = blockIdx.x / blocksPerBatch;
  const int t = (blockIdx.x % blocksPerBatch) * TPB_TOPK + (int)threadIdx.x;

  const float* sb = scores + b * T;
  for (int i = threadIdx.x; i < T; i += TPB_TOPK) s[i] = sb[i];
  __syncthreads();

  const float mine = s[t];
  int rank = 0;
#pragma unroll 8
  for (int j = 0; j < T; ++j) {
    const float v = s[j];                     // LDS broadcast: no bank conflict
    rank += (v > mine) || (v == mine && j < t);
  }
  if (rank < K) idx_out[b * K + rank] = t;
}

// ---------------------------------------------------------------------------
// Kernel 3: gather selected rows, one 8 KB row per block, float4 coalesced.
// ---------------------------------------------------------------------------
__global__ void __launch_bounds__(256) mod_gather(
    const float* __restrict__ x, const int* __restrict__ idx,
    float* __restrict__ out)
{
  const int r = blockIdx.x;                   // 0 .. B*K-1
  const int b = r / K;
  const int t = idx[r];
  const float4* __restrict__ src = (const float4*)(x + ((size_t)b * T + t) * D);
  float4* __restrict__ dst       = (float4*)(out + (size_t)r * D);
#pragma unroll
  for (int i = threadIdx.x; i < D / 4; i += 256) dst[i] = src[i];
}

// ---------------------------------------------------------------------------
// d_out layout (tuple concat, flat):
//   [0,            B*K*D)                     selected_x  (float)
//   [B*K*D,        B*K*D + B*K)               indices     (int32 bits)
//   [B*K*D + B*K,  B*K*D + B*K + B*T)         router_scores (float)
// ---------------------------------------------------------------------------
extern "C" void kernel_launch(void* const* d_in, const int* in_sizes, int n_in,
                              void* d_out, int out_size, void* d_ws, size_t ws_size,
                              hipStream_t stream) {
  const float* x = (const float*)d_in[0];
  const float* w = (const float*)d_in[1];

  float* out    = (float*)d_out;
  float* sel    = out;
  int*   idx    = (int*)(out + (size_t)B * K * D);
  float* scores = out + (size_t)B * K * D + (size_t)B * K;

  // 256 blocks x 8 waves x 16 tokens = B*T tokens
  mod_scores_wmma<<<(B * T) / 128, 256, 0, stream>>>(x, w, scores);
  mod_topk_rank<<<B * (T / TPB_TOPK), TPB_TOPK, 0, stream>>>(scores, idx);
  mod_gather<<<B * K, 256, 0, stream>>>(x, idx, sel);
}